// att_net_23751169147015
// MI455X (gfx1250) — compile-verified
//
#include <hip/hip_runtime.h>
#include <math.h>

typedef float v2f __attribute__((ext_vector_type(2)));
typedef float v8f __attribute__((ext_vector_type(8)));

#define IN_DIM 256   // row width of every activation matrix in this net
#define HID    128   // per-branch GEMM output width
#define NCLS   7
#define NB     512
#define LDSPAD 260   // 256 + 4 pad -> conflict-free LDS reads (bank = (4m+k)&63)

// ---------------------------------------------------------------------------
// Fused gather + mean over S sampled neighbors.
//   out[row][c] = (1/S) * sum_j src[ idx ? idx[row*S+j] : row*S+j ][c]
// One 256-thread block per output row; each j-step reads one fully coalesced
// 1KB row. This is the HBM-bound stage (~131MB for f2).
// ---------------------------------------------------------------------------
__global__ void mean_gather(float* __restrict__ out, const float* __restrict__ src,
                            const int* __restrict__ idx, int S, float invS) {
  const int row = blockIdx.x;
  const int col = threadIdx.x;              // 0..255
  const long base = (long)row * S;
  float acc = 0.f;
  for (int j = 0; j < S; ++j) {
    const int r = idx ? idx[base + j] : (int)(base + j);
    acc += src[(long)r * IN_DIM + col];
  }
  out[(long)row * IN_DIM + col] = acc * invS;
}

// ---------------------------------------------------------------------------
// C[M,HID] = act(A[M,256] @ W[256,HID] + bias), A rows optionally gathered
// through aIdx. Full-f32 WMMA: V_WMMA_F32_16X16X4_F32, K accumulated in f32.
// Block = 256 threads = 8 waves; block owns one 16-row M tile staged in LDS;
// wave w owns output columns [16w, 16w+16).
// A operand layout (32-bit A 16x4): lanes 0-15 -> K=k,k+1 of row M=lane;
// lanes 16-31 -> K=k+2,k+3. B (4x16): N=lane&15, same K split.
// C/D: VGPR v -> M = v + 8*(lane>=16), N = lane&15.
// ---------------------------------------------------------------------------
__global__ void gemm256x128_wmma(const float* __restrict__ A, const int* __restrict__ aIdx,
                                 const float* __restrict__ W, const float* __restrict__ bias,
                                 float* __restrict__ out, int ldc, int doRelu) {
  __shared__ float sA[16 * LDSPAD];
  const int tile = blockIdx.x;
  const int t = threadIdx.x;                // 0..255

  // Stage the 16x256 A tile (each i loads one full row, perfectly coalesced).
  for (int i = 0; i < 16; ++i) {
    const int gr = tile * 16 + i;
    const int r = aIdx ? aIdx[gr] : gr;
    sA[i * LDSPAD + t] = A[(long)r * IN_DIM + t];
  }
  __syncthreads();

  const int w    = t >> 5;                  // wave id -> N tile
  const int lane = t & 31;
  const int hi   = lane >> 4;               // upper half-wave handles K+2,K+3
  const int m    = lane & 15;
  const int n    = (w << 4) + (lane & 15);

  const float* sa = &sA[m * LDSPAD + 2 * hi];
  const float* wb = &W[(2 * hi) * HID + n];

  v8f c = {};
#pragma unroll 8
  for (int k = 0; k < IN_DIM; k += 4) {
    v2f a, b;
    a.x = sa[k];
    a.y = sa[k + 1];
    b.x = wb[(long)k * HID];
    b.y = wb[(long)(k + 1) * HID];
    // D = A(16x4) * B(4x16) + C, full fp32
    c = __builtin_amdgcn_wmma_f32_16x16x4_f32(false, a, false, b,
                                              (short)0, c, false, false);
  }

  const float bv = bias[n];
  const int rowBase = tile * 16 + 8 * hi;
#pragma unroll
  for (int v = 0; v < 8; ++v) {
    float val = c[v] + bv;
    if (doRelu) val = fmaxf(val, 0.f);
    out[(long)(rowBase + v) * ldc + n] = val;
  }
}

// ---------------------------------------------------------------------------
// out[b] = (g0[b] / max(||g0[b]||, 1e-12)) @ W_fc + b_fc
// One wave32 per row; shuffle reductions (norm is a per-row scalar, so we
// compute raw dots and scale once).
// ---------------------------------------------------------------------------
__global__ void norm_fc(const float* __restrict__ g0, const float* __restrict__ Wfc,
                        const float* __restrict__ bfc, float* __restrict__ out) {
  const int row  = (blockIdx.x * blockDim.x + threadIdx.x) >> 5; // 512 waves exactly
  const int lane = threadIdx.x & 31;
  const float* gr = g0 + (long)row * IN_DIM;

  float ss = 0.f;
  float pc[NCLS];
#pragma unroll
  for (int cc = 0; cc < NCLS; ++cc) pc[cc] = 0.f;

#pragma unroll
  for (int i = 0; i < 8; ++i) {
    const int k = lane + 32 * i;
    const float x = gr[k];
    ss += x * x;
    const float* wr = &Wfc[k * NCLS];
#pragma unroll
    for (int cc = 0; cc < NCLS; ++cc) pc[cc] += x * wr[cc];
  }
  for (int off = 16; off >= 1; off >>= 1) {
    ss += __shfl_xor(ss, off, 32);
#pragma unroll
    for (int cc = 0; cc < NCLS; ++cc) pc[cc] += __shfl_xor(pc[cc], off, 32);
  }
  if (lane == 0) {
    const float inv = 1.0f / fmaxf(sqrtf(ss), 1e-12f);
#pragma unroll
    for (int cc = 0; cc < NCLS; ++cc)
      out[(long)row * NCLS + cc] = pc[cc] * inv + bfc[cc];
  }
}

// ---------------------------------------------------------------------------
extern "C" void kernel_launch(void* const* d_in, const int* in_sizes, int n_in,
                              void* d_out, int out_size, void* d_ws, size_t ws_size,
                              hipStream_t stream) {
  (void)in_sizes; (void)n_in; (void)out_size; (void)ws_size;

  const int*   ids   = (const int*)  d_in[0];
  const int*   ids1  = (const int*)  d_in[1];
  const int*   ids2  = (const int*)  d_in[2];
  const float* feats = (const float*)d_in[3];
  const float* Wx1   = (const float*)d_in[4];
  const float* bx1   = (const float*)d_in[5];
  const float* Wn1   = (const float*)d_in[6];
  const float* bn1   = (const float*)d_in[7];
  const float* Wx2   = (const float*)d_in[8];
  const float* bx2   = (const float*)d_in[9];
  const float* Wn2   = (const float*)d_in[10];
  const float* bn2   = (const float*)d_in[11];
  const float* Wfc   = (const float*)d_in[12];
  const float* bfc   = (const float*)d_in[13];

  float* ws  = (float*)d_ws;
  float* m2  = ws;                               // [12800,256] mean of f2 groups
  float* m1  = m2  + (size_t)12800 * 256;        // [512,256]   mean of f1 groups
  float* h1  = m1  + (size_t)512   * 256;        // [12800,256] layer-1 out (hop1)
  float* h0  = h1  + (size_t)12800 * 256;        // [512,256]   layer-1 out (hop0)
  float* mh1 = h0  + (size_t)512   * 256;        // [512,256]   mean of h1 groups
  float* g0  = mh1 + (size_t)512   * 256;        // [512,256]   layer-2 out

  // neighbor means (HBM-bound gather stage)
  mean_gather<<<12800, 256, 0, stream>>>(m2, feats, ids2, 10, 0.1f);
  mean_gather<<<512,   256, 0, stream>>>(m1, feats, ids1, 25, 1.f / 25.f);

  // layer 1 (ReLU fused):  h = relu([x@Wx1+bx1 | m@Wn1+bn1])
  gemm256x128_wmma<<<800, 256, 0, stream>>>(feats, ids1, Wx1, bx1, h1,       IN_DIM, 1);
  gemm256x128_wmma<<<800, 256, 0, stream>>>(m2,  nullptr, Wn1, bn1, h1 + HID, IN_DIM, 1);
  gemm256x128_wmma<<<32,  256, 0, stream>>>(feats, ids,  Wx1, bx1, h0,       IN_DIM, 1);
  gemm256x128_wmma<<<32,  256, 0, stream>>>(m1,  nullptr, Wn1, bn1, h0 + HID, IN_DIM, 1);

  // layer 2 (no activation)
  mean_gather<<<512, 256, 0, stream>>>(mh1, h1, nullptr, 25, 1.f / 25.f);
  gemm256x128_wmma<<<32, 256, 0, stream>>>(h0,  nullptr, Wx2, bx2, g0,       IN_DIM, 0);
  gemm256x128_wmma<<<32, 256, 0, stream>>>(mh1, nullptr, Wn2, bn2, g0 + HID, IN_DIM, 0);

  // row-normalize + classifier
  norm_fc<<<64, 256, 0, stream>>>(g0, Wfc, bfc, (float*)d_out);
}